// JointBilateralFilter_45930380264075
// MI455X (gfx1250) — compile-verified
//
#include <hip/hip_runtime.h>
#include <hip/hip_fp16.h>

// Joint bilateral filter (K=5), N=2, guide C=3, flow Cv=2, H=720, W=1280.
// coeff = clip(1 - |NEG_SS_DIV - idiff*SI_DIV|, 0, 1) == max(0, 0.875 - 50*idiff)
// Strength-reduced: idiff = qc + qs - 2*tc.s =>
//   cf = (0.875 - 50*qc) + (-50*qs) + 100*(tc.s)
// with q50 = -50*qs precomputed per halo pixel in a 6th LDS plane.

#define KSZ     5
#define HALO    2
#define TILE_W  64
#define TILE_H  16
#define LW      (TILE_W + 2 * HALO)   // 68
#define LH      (TILE_H + 2 * HALO)   // 20
#define PLANE   (LW * LH)             // 1360 floats
#define NSTAGE  5                     // 3 guide + 2 flow planes staged from memory
#define STAGE_ELEMS (PLANE * NSTAGE)  // 6800 floats
#define SMEM_ELEMS  (PLANE * 6)       // + q50 plane = 8160 floats = 32.6 KB

#define IMG_H   720
#define IMG_W   1280
#define NBATCH  2
#define NTHREADS 256

typedef float f32x2 __attribute__((ext_vector_type(2)));
typedef float f32x4 __attribute__((ext_vector_type(4)));

static __device__ __forceinline__ f32x2 splat2(float v) {
    f32x2 r; r.x = v; r.y = v; return r;
}
static __device__ __forceinline__ f32x2 pack2(float a, float b) {
    f32x2 r; r.x = a; r.y = b; return r;
}
// 8 contiguous LDS floats, base index multiple of 4 -> two ds_load_b128.
static __device__ __forceinline__ void ldrow8(const float* p, float* d) {
    const f32x4 u = *(const f32x4*)(p);
    const f32x4 w = *(const f32x4*)(p + 4);
    d[0] = u.x; d[1] = u.y; d[2] = u.z; d[3] = u.w;
    d[4] = w.x; d[5] = w.y; d[6] = w.z; d[7] = w.w;
}

__global__ __launch_bounds__(NTHREADS) void
JointBilateralFilter_45930380264075_kernel(const float* __restrict__ t,
                                           const float* __restrict__ v,
                                           __half* __restrict__ out)
{
    __shared__ __align__(16) float sm[SMEM_ELEMS];
    float* __restrict__ smq = sm + NSTAGE * PLANE;   // q50 plane

    const int tid = threadIdx.x;
    const int bx = blockIdx.x, by = blockIdx.y, n = blockIdx.z;
    const int gx0 = bx * TILE_W;
    const int gy0 = by * TILE_H;

    // Blocks whose halo touches the image border need zero padding in LDS.
    const bool border = (gx0 < HALO) || (gy0 < HALO) ||
                        (gx0 + TILE_W + HALO > IMG_W) ||
                        (gy0 + TILE_H + HALO > IMG_H);
    if (border) {
        f32x4* smv = (f32x4*)sm;
        const f32x4 z4 = {0.0f, 0.0f, 0.0f, 0.0f};
        for (int i = tid; i < STAGE_ELEMS / 4; i += NTHREADS) smv[i] = z4;
        __syncthreads();   // zeros visible before async writes land on top
    }

    // ---- Stage halo tile (5 planes) with CDNA5 async global->LDS b64 copies.
    //      Chunks are 2 floats at even columns: 8B-aligned, and always fully
    //      in- or out-of-bounds at image edges (start even, width 2). ----
    const float* tbase = t + (size_t)n * 3 * IMG_H * IMG_W;
    const float* vbase = v + (size_t)n * 2 * IMG_H * IMG_W;

    const int NCHUNK = STAGE_ELEMS / 2;        // 3400 b64 chunks
    for (int k = tid; k < NCHUNK; k += NTHREADS) {
        const int plane = k / (PLANE / 2);     // /680
        const int rem   = k - plane * (PLANE / 2);
        const int r     = rem / (LW / 2);      // /34
        const int c2    = rem - r * (LW / 2);
        const int gy    = gy0 - HALO + r;
        const int gx    = gx0 - HALO + 2 * c2; // even
        if ((unsigned)gy < (unsigned)IMG_H && (unsigned)gx < (unsigned)IMG_W) {
            const float* src = (plane < 3)
                ? (tbase + ((size_t)plane * IMG_H + gy) * IMG_W + gx)
                : (vbase + ((size_t)(plane - 3) * IMG_H + gy) * IMG_W + gx);
            // Low 32 bits of a flat pointer into LDS == LDS byte offset.
            unsigned lds_off = (unsigned)(unsigned long long)
                               (const void*)(sm + plane * PLANE + r * LW + 2 * c2);
            unsigned long long ga = (unsigned long long)src;
            asm volatile("global_load_async_to_lds_b64 %0, %1, off"
                         :: "v"(lds_off), "v"(ga) : "memory");
        }
    }
    asm volatile("s_wait_asynccnt 0" ::: "memory");
    __syncthreads();

    // ---- q50 plane: -50 * sum_c guide^2, once per halo pixel (vectorized) ----
    {
        const f32x4* g0v = (const f32x4*)sm;
        const f32x4* g1v = (const f32x4*)(sm + PLANE);
        const f32x4* g2v = (const f32x4*)(sm + 2 * PLANE);
        f32x4* qv = (f32x4*)smq;
        for (int i = tid; i < PLANE / 4; i += NTHREADS) {
            const f32x4 g0 = g0v[i], g1 = g1v[i], g2 = g2v[i];
            f32x4 q = __builtin_elementwise_fma(g2, g2,
                          __builtin_elementwise_fma(g1, g1, g0 * g0));
            const f32x4 m50 = {-50.0f, -50.0f, -50.0f, -50.0f};
            qv[i] = q * m50;
        }
    }
    __syncthreads();

    // ---- Compute: 4 horizontally adjacent pixels per thread,
    //      as two 2-wide packed-f32 pixel pairs (A: x0..x0+1, B: x0+2..x0+3) ----
    const int qx = tid & 15;          // quad index, 0..15
    const int ly = tid >> 4;          // row in tile, 0..15
    const int x0 = qx * 4;            // tile-local x of leftmost pixel (mult of 4)
    const int cb = (ly + HALO) * LW + (x0 + HALO);   // center of pixel x0

    const f32x2 tcA0 = pack2(sm[cb],                 sm[cb + 1]);
    const f32x2 tcA1 = pack2(sm[PLANE + cb],         sm[PLANE + cb + 1]);
    const f32x2 tcA2 = pack2(sm[2 * PLANE + cb],     sm[2 * PLANE + cb + 1]);
    const f32x2 tcB0 = pack2(sm[cb + 2],             sm[cb + 3]);
    const f32x2 tcB1 = pack2(sm[PLANE + cb + 2],     sm[PLANE + cb + 3]);
    const f32x2 tcB2 = pack2(sm[2 * PLANE + cb + 2], sm[2 * PLANE + cb + 3]);

    const f32x2 k100 = splat2(100.0f);
    const f32x2 wA0 = tcA0 * k100, wA1 = tcA1 * k100, wA2 = tcA2 * k100;
    const f32x2 wB0 = tcB0 * k100, wB1 = tcB1 * k100, wB2 = tcB2 * k100;
    const f32x2 kC = splat2(0.875f);
    const f32x2 C0A = kC + pack2(smq[cb],     smq[cb + 1]);
    const f32x2 C0B = kC + pack2(smq[cb + 2], smq[cb + 3]);
    const f32x2 kZero = splat2(0.0f);

    f32x2 numA0 = splat2(0.0f), numA1 = splat2(0.0f), denA = splat2(0.0f);
    f32x2 numB0 = splat2(0.0f), numB1 = splat2(0.0f), denB = splat2(0.0f);

#pragma unroll
    for (int dy = 0; dy < KSZ; ++dy) {
        const int ro = (ly + dy) * LW + x0;   // mult of 4 -> 16B aligned
        float a0[8], a1[8], a2[8], b0[8], b1[8], qr[8];
        ldrow8(sm + ro,             a0);
        ldrow8(sm + PLANE + ro,     a1);
        ldrow8(sm + 2 * PLANE + ro, a2);
        ldrow8(sm + 3 * PLANE + ro, b0);
        ldrow8(sm + 4 * PLANE + ro, b1);
        ldrow8(smq + ro,            qr);
#pragma unroll
        for (int dx = 0; dx < KSZ; ++dx) {
            // pair A (pixels x0, x0+1): columns dx, dx+1
            f32x2 accA = C0A + pack2(qr[dx], qr[dx + 1]);
            accA = __builtin_elementwise_fma(wA0, pack2(a0[dx], a0[dx + 1]), accA);
            accA = __builtin_elementwise_fma(wA1, pack2(a1[dx], a1[dx + 1]), accA);
            accA = __builtin_elementwise_fma(wA2, pack2(a2[dx], a2[dx + 1]), accA);
            const f32x2 cfA = __builtin_elementwise_max(accA, kZero);
            numA0 = __builtin_elementwise_fma(pack2(b0[dx], b0[dx + 1]), cfA, numA0);
            numA1 = __builtin_elementwise_fma(pack2(b1[dx], b1[dx + 1]), cfA, numA1);
            denA = denA + cfA;
            // pair B (pixels x0+2, x0+3): columns dx+2, dx+3
            f32x2 accB = C0B + pack2(qr[dx + 2], qr[dx + 3]);
            accB = __builtin_elementwise_fma(wB0, pack2(a0[dx + 2], a0[dx + 3]), accB);
            accB = __builtin_elementwise_fma(wB1, pack2(a1[dx + 2], a1[dx + 3]), accB);
            accB = __builtin_elementwise_fma(wB2, pack2(a2[dx + 2], a2[dx + 3]), accB);
            const f32x2 cfB = __builtin_elementwise_max(accB, kZero);
            numB0 = __builtin_elementwise_fma(pack2(b0[dx + 2], b0[dx + 3]), cfB, numB0);
            numB1 = __builtin_elementwise_fma(pack2(b1[dx + 2], b1[dx + 3]), cfB, numB1);
            denB = denB + cfB;
        }
    }

    // den >= 0.875 (center tap has cf == 0.875), rcp is safe; output is fp16.
    const f32x2 rA = pack2(__builtin_amdgcn_rcpf(denA.x), __builtin_amdgcn_rcpf(denA.y));
    const f32x2 rB = pack2(__builtin_amdgcn_rcpf(denB.x), __builtin_amdgcn_rcpf(denB.y));
    const f32x2 oA0 = numA0 * rA, oA1 = numA1 * rA;
    const f32x2 oB0 = numB0 * rB, oB1 = numB1 * rB;

    const int ox = gx0 + x0;          // mult of 4 -> 8B-aligned half stores
    const int oy = gy0 + ly;
    const size_t oidx = (((size_t)n * 2 + 0) * IMG_H + oy) * IMG_W + ox;

    float2 p0, p1;
    ((__half2*)&p0)[0] = __floats2half2_rn(oA0.x, oA0.y);
    ((__half2*)&p0)[1] = __floats2half2_rn(oB0.x, oB0.y);
    ((__half2*)&p1)[0] = __floats2half2_rn(oA1.x, oA1.y);
    ((__half2*)&p1)[1] = __floats2half2_rn(oB1.x, oB1.y);
    *(float2*)(out + oidx) = p0;                            // 4 halves, one b64 store
    *(float2*)(out + oidx + (size_t)IMG_H * IMG_W) = p1;    // 4 halves, one b64 store
}

extern "C" void kernel_launch(void* const* d_in, const int* in_sizes, int n_in,
                              void* d_out, int out_size, void* d_ws, size_t ws_size,
                              hipStream_t stream) {
    (void)in_sizes; (void)n_in; (void)out_size; (void)d_ws; (void)ws_size;
    const float* t = (const float*)d_in[0];        // (2,3,720,1280) f32
    const float* v = (const float*)d_in[1];        // (2,2,720,1280) f32
    __half* out = (__half*)d_out;                  // (2,2,720,1280) f16

    dim3 grid(IMG_W / TILE_W, IMG_H / TILE_H, NBATCH);  // (20, 45, 2)
    JointBilateralFilter_45930380264075_kernel<<<grid, NTHREADS, 0, stream>>>(t, v, out);
}